// GQA_22436909154699
// MI455X (gfx1250) — compile-verified
//
#include <hip/hip_runtime.h>

// GQA reference reduces exactly to: out[b,l,g,h,:] = (x @ Wv_g + bv_g)  (softmax over
// a singleton axis == 1, so q/Wq/bq are dead).  We compute the V projection with
// fp32 WMMA (V_WMMA_F32_16X16X4_F32) and broadcast-store with NT hints.

typedef float v2f __attribute__((ext_vector_type(2)));
typedef float v8f __attribute__((ext_vector_type(8)));

#define K_DIM      2048   // embed dim (GEMM K)
#define ROW_OUT    2048   // output row width
#define KV_W       512    // Wkv row width
#define N_V        256    // total v columns (4 groups * 64)
#define M_TILE     64     // rows of x per block
#define KC         64     // K chunk staged in LDS
#define LDS_STRIDE 68     // 68*4B = 272B: multiple of 16B, conflict-free b64 reads

__global__ __launch_bounds__(512)
void gqa_vproj_wmma(const float* __restrict__ x,
                    const float* __restrict__ Wkv,
                    const float* __restrict__ bkv,
                    float* __restrict__ out)
{
    __shared__ float ldsA[M_TILE * LDS_STRIDE];

    const int tid   = threadIdx.x;
    const int lane  = tid & 31;
    const int wave  = tid >> 5;            // 0..15 -> one 16-wide N tile each
    const int m0    = blockIdx.x * M_TILE;

    const int n0    = wave * 16;           // column tile in compacted V space
    const int g     = n0 >> 6;             // group id (tile never crosses a group)
    const int d0    = n0 & 63;             // head-dim base
    const int col0  = g * 128 + 64 + d0;   // v-columns of group g inside Wkv

    const int nlane = lane & 15;           // N within tile (B/C/D layout)
    const int hi    = lane >> 4;           // 0 -> K pair (0,1), 1 -> K pair (2,3)

    v8f acc[4];
    #pragma unroll
    for (int i = 0; i < 4; ++i) acc[i] = (v8f){};

    for (int kc = 0; kc < K_DIM; kc += KC) {
        // ---- cooperative, coalesced stage of the 64x64 A chunk into LDS ----
        #pragma unroll
        for (int r = 0; r < 2; ++r) {
            const int row = r * 32 + (tid >> 4);
            const int c4  = (tid & 15) * 4;
            const size_t gsrc = (size_t)(m0 + row) * K_DIM + kc + c4;
            const float4 v = *(const float4*)(x + gsrc);
            *(float4*)(ldsA + row * LDS_STRIDE + c4) = v;
            if (kc + KC < K_DIM)   // hint next chunk into cache (global_prefetch_b8)
                __builtin_prefetch(x + gsrc + KC, 0, 3);
        }
        __syncthreads();

        // ---- 16 K-steps of 4; 4 M-subtiles per wave per step ----
        #pragma unroll
        for (int kk = 0; kk < KC; kk += 4) {
            const int krow = kc + kk + 2 * hi;    // ISA A/B fp32 K-pair layout
            v2f bfrag;
            bfrag.x = Wkv[(size_t)(krow + 0) * KV_W + col0 + nlane];
            bfrag.y = Wkv[(size_t)(krow + 1) * KV_W + col0 + nlane];

            #pragma unroll
            for (int ms = 0; ms < 4; ++ms) {
                const int aidx = (ms * 16 + nlane) * LDS_STRIDE + kk + 2 * hi;
                v2f afrag = *(const v2f*)(ldsA + aidx);
                acc[ms] = __builtin_amdgcn_wmma_f32_16x16x4_f32(
                    /*neg_a=*/false, afrag, /*neg_b=*/false, bfrag,
                    /*c_mod=*/(short)0, acc[ms],
                    /*reuse_a=*/false, /*reuse_b=*/false);
            }
        }
        __syncthreads();
    }

    // ---- epilogue: bias add, broadcast each v over the 8 heads of its group ----
    const float bb = bkv[col0 + nlane];
    #pragma unroll
    for (int ms = 0; ms < 4; ++ms) {
        #pragma unroll
        for (int i = 0; i < 8; ++i) {
            const int   m    = m0 + ms * 16 + i + 8 * hi;  // C/D layout: vgpr i, half `hi`
            const float vv   = acc[ms][i] + bb;
            const size_t ob  = (size_t)m * ROW_OUT + (size_t)g * 512 + d0 + nlane;
            #pragma unroll
            for (int h = 0; h < 8; ++h)
                __builtin_nontemporal_store(vv, out + ob + h * 64);  // write-once stream
        }
    }
}

extern "C" void kernel_launch(void* const* d_in, const int* in_sizes, int n_in,
                              void* d_out, int out_size, void* d_ws, size_t ws_size,
                              hipStream_t stream) {
    (void)in_sizes; (void)n_in; (void)d_ws; (void)ws_size; (void)out_size;
    const float* x   = (const float*)d_in[0];
    // d_in[1] (Wq) and d_in[2] (bq) are mathematically dead: softmax over a
    // singleton axis is identically 1, so the output is exactly the broadcast V.
    const float* Wkv = (const float*)d_in[3];
    const float* bkv = (const float*)d_in[4];
    float* out = (float*)d_out;

    const int M = 4 * 4096;                 // B * L
    dim3 grid(M / M_TILE), block(512);
    gqa_vproj_wmma<<<grid, block, 0, stream>>>(x, Wkv, bkv, out);
}